// GaussianVoxelizer_72060961292852
// MI455X (gfx1250) — compile-verified
//
#include <hip/hip_runtime.h>
#include <hip/hip_bf16.h>

// ---------------- problem constants (from reference) ----------------
#define HH 100
#define WW2 100
#define DD 8
#define PTS (HH * WW2 * DD)        // 80000
#define NCLS 18
#define CPAD 32                    // classes padded to two 16-wide WMMA tiles
#define NG 1024
#define NPAD 1032                  // 1025 gaussians padded (multiple of 8)
#define PSTRIDE 16                 // floats per gaussian param record (64 B)

typedef float v2f __attribute__((ext_vector_type(2)));
typedef float v8f __attribute__((ext_vector_type(8)));

// ---------------- prep: per-gaussian Sigma^-1, mu, box, opacity, padded feats ----------
__global__ void gv_prep(const float* __restrict__ means3d,
                        const float* __restrict__ opac,
                        const float* __restrict__ scales,
                        const float* __restrict__ rots,
                        const float* __restrict__ features,
                        const float* __restrict__ empty_scalar,
                        float* __restrict__ params,
                        float* __restrict__ featsPad) {
  int n = blockIdx.x * blockDim.x + threadIdx.x;
  if (n >= NPAD) return;

  float c00, c01, c02, c11, c12, c22, mux, muy, muz, rx, ry, rz, op;
  if (n < NG) {
    float qw = rots[4 * n + 0], qx = rots[4 * n + 1];
    float qy = rots[4 * n + 2], qz = rots[4 * n + 3];
    float inv = rsqrtf(qw * qw + qx * qx + qy * qy + qz * qz);
    qw *= inv; qx *= inv; qy *= inv; qz *= inv;
    float r00 = 1.f - 2.f * (qy * qy + qz * qz);
    float r01 = 2.f * (qx * qy - qw * qz);
    float r02 = 2.f * (qx * qz + qw * qy);
    float r10 = 2.f * (qx * qy + qw * qz);
    float r11 = 1.f - 2.f * (qx * qx + qz * qz);
    float r12 = 2.f * (qy * qz - qw * qx);
    float r20 = 2.f * (qx * qz - qw * qy);
    float r21 = 2.f * (qy * qz + qw * qx);
    float r22 = 1.f - 2.f * (qx * qx + qy * qy);
    float sx = scales[3 * n + 0], sy = scales[3 * n + 1], sz = scales[3 * n + 2];
    float s0 = sx * sx, s1 = sy * sy, s2 = sz * sz;
    c00 = r00 * r00 * s0 + r01 * r01 * s1 + r02 * r02 * s2;
    c01 = r00 * r10 * s0 + r01 * r11 * s1 + r02 * r12 * s2;
    c02 = r00 * r20 * s0 + r01 * r21 * s1 + r02 * r22 * s2;
    c11 = r10 * r10 * s0 + r11 * r11 * s1 + r12 * r12 * s2;
    c12 = r10 * r20 * s0 + r11 * r21 * s1 + r12 * r22 * s2;
    c22 = r20 * r20 * s0 + r21 * r21 * s1 + r22 * r22 * s2;
    mux = means3d[3 * n + 0]; muy = means3d[3 * n + 1]; muz = means3d[3 * n + 2];
    rx = 3.f * sx; ry = 3.f * sy; rz = 3.f * sz;
    op = opac[n];
  } else if (n == NG) {
    // empty gaussian: Sigma = diag(rng^2), rng = (80,80,6.4), mu = center
    c00 = 6400.f; c01 = 0.f; c02 = 0.f; c11 = 6400.f; c12 = 0.f; c22 = 40.96f;
    mux = 0.f; muy = 0.f; muz = 2.2f;
    rx = 240.f; ry = 240.f; rz = 19.2f;
    op = 1.f;
  } else {
    // padding rows: never culled-in (mu far away), weight forced to zero anyway
    c00 = 1.f; c01 = 0.f; c02 = 0.f; c11 = 1.f; c12 = 0.f; c22 = 1.f;
    mux = muy = muz = 1.0e9f;
    rx = ry = rz = -1.f;
    op = 0.f;
  }

  // symmetric 3x3 inverse via adjugate
  float m00 = c11 * c22 - c12 * c12;
  float m01 = c02 * c12 - c01 * c22;
  float m02 = c01 * c12 - c02 * c11;
  float det = c00 * m00 + c01 * m01 + c02 * m02;
  float id = 1.f / det;
  float a00 = m00 * id;
  float a01 = m01 * id;
  float a02 = m02 * id;
  float a11 = (c00 * c22 - c02 * c02) * id;
  float a12 = (c01 * c02 - c00 * c12) * id;
  float a22 = (c00 * c11 - c01 * c01) * id;

  // 64-byte record: {mux,muy,muz,op | rx,ry,rz,a00 | a01,a02,a11,a12 | a22,0,0,0}
  float* pg = params + n * PSTRIDE;
  pg[0] = mux; pg[1] = muy; pg[2] = muz; pg[3] = op;
  pg[4] = rx;  pg[5] = ry;  pg[6] = rz;  pg[7] = a00;
  pg[8] = a01; pg[9] = a02; pg[10] = a11; pg[11] = a12;
  pg[12] = a22; pg[13] = 0.f; pg[14] = 0.f; pg[15] = 0.f;

  // plain padded feats: featsPad[n*32 + c]
  float* fp = featsPad + n * CPAD;
  for (int c = 0; c < CPAD; ++c) {
    float v = 0.f;
    if (n < NG && c < NCLS - 1) v = features[n * (NCLS - 1) + c];
    if (n == NG && c == NCLS - 1) v = empty_scalar[0];
    fp[c] = v;
  }
}

// ---------------- main: deterministic cull/compact -> WMMA-f32 splat loop --------------
__global__ __launch_bounds__(256) void gv_main(const float* __restrict__ params,
                                               const float* __restrict__ feats,
                                               float* __restrict__ out) {
  __shared__ int s_list[NPAD + 8];
  __shared__ int s_wc[8];

  const int lane = threadIdx.x & 31;
  const int wv = threadIdx.x >> 5;

  // ---- Phase 1: ordered (deterministic) cull against this block's 128-point AABB ----
  const int B0 = blockIdx.x * 128;
  const int hmin = B0 / (WW2 * DD);
  const int hmax = (B0 + 127) / (WW2 * DD);
  int wmin = 0, wmax = WW2 - 1;
  if (hmin == hmax) {
    wmin = (B0 % (WW2 * DD)) >> 3;
    wmax = ((B0 + 127) % (WW2 * DD)) >> 3;
  }
  const float xlo = ((float)hmin + 0.5f) * 0.8f - 40.f;
  const float xhi = ((float)hmax + 0.5f) * 0.8f - 40.f;
  const float ylo = ((float)wmin + 0.5f) * 0.8f - 40.f;
  const float yhi = ((float)wmax + 0.5f) * 0.8f - 40.f;
  const float zlo = -0.6f;  // d=0
  const float zhi = 5.0f;   // d=7

  int total = 0;  // block-uniform running count -> list sorted by gaussian index
  for (int base = 0; base < NPAD; base += 256) {
    const int n = base + threadIdx.x;
    int pass = 0;
    if (n < NPAD) {
      const float4* rg = (const float4*)(params + n * PSTRIDE);
      float4 F0 = rg[0];  // mux,muy,muz,op
      float4 F1 = rg[1];  // rx, ry, rz, a00
      pass = (F0.x >= xlo - F1.x) & (F0.x <= xhi + F1.x) &
             (F0.y >= ylo - F1.y) & (F0.y <= yhi + F1.y) &
             (F0.z >= zlo - F1.z) & (F0.z <= zhi + F1.z);
    }
    unsigned mb = __builtin_amdgcn_ballot_w32(pass != 0);
    int prefix = __builtin_popcount(mb & ((1u << lane) - 1u));
    if (lane == 0) s_wc[wv] = __builtin_popcount(mb);
    __syncthreads();
    int woff = 0, chunkTotal = 0;
#pragma unroll
    for (int i = 0; i < 8; ++i) {
      int c = s_wc[i];
      if (i < wv) woff += c;
      chunkTotal += c;
    }
    if (pass) s_list[total + woff + prefix] = n;
    total += chunkTotal;
    __syncthreads();
  }

  const int nc = total;
  const int ncp = (nc + 3) & ~3;
  if (threadIdx.x < (ncp - nc)) s_list[nc + threadIdx.x] = NPAD - 1;  // dead record pad
  __syncthreads();

  // ---- Phase 2: WMMA accumulation over the compacted list ----
  const int m = lane & 15;        // point within the wave's 16-point tile
  const int hl = lane >> 4;       // which K-pair this lane owns (A-matrix layout)
  const int tileBase = B0 + wv * 16;

  const int p = tileBase + m;
  const int h = p / (WW2 * DD);
  const int rem = p - h * (WW2 * DD);
  const int w = rem >> 3;
  const int d = rem & 7;
  const float x0 = ((float)h + 0.5f) * 0.8f - 40.f;
  const float x1 = ((float)w + 0.5f) * 0.8f - 40.f;
  const float x2 = ((float)d + 0.5f) * 0.8f - 1.f;

  v8f acc0 = {0.f, 0.f, 0.f, 0.f, 0.f, 0.f, 0.f, 0.f};  // classes 0..15
  v8f acc1 = {0.f, 0.f, 0.f, 0.f, 0.f, 0.f, 0.f, 0.f};  // classes 16..31

#pragma unroll 2
  for (int j = 0; j < ncp; j += 4) {
    const int g0 = j + hl * 2;  // this lane evaluates list slots g0, g0+1 (g0 even)
    int2 ip = *(const int2*)(s_list + g0);  // ds_load_b64

    const float4* ra = (const float4*)(params + ip.x * PSTRIDE);
    const float4* rb = (const float4*)(params + ip.y * PSTRIDE);
    float4 Fa0 = ra[0], Fa1 = ra[1], Fa2 = ra[2], Fa3 = ra[3];
    float4 Fb0 = rb[0], Fb1 = rb[1], Fb2 = rb[2], Fb3 = rb[3];

    // branchless: exp kept live in all lanes (0*exp not foldable), opa gated by mask
    float dxa = x0 - Fa0.x, dya = x1 - Fa0.y, dza = x2 - Fa0.z;
    int inA = (fabsf(dxa) <= Fa1.x) & (fabsf(dya) <= Fa1.y) & (fabsf(dza) <= Fa1.z);
    float qa = Fa1.w * dxa * dxa + Fa2.z * dya * dya + Fa3.x * dza * dza +
               2.f * (Fa2.x * dxa * dya + Fa2.y * dxa * dza + Fa2.w * dya * dza);
    float wa = (inA ? Fa0.w : 0.f) * __expf(-0.5f * qa);

    float dxb = x0 - Fb0.x, dyb = x1 - Fb0.y, dzb = x2 - Fb0.z;
    int inB = (fabsf(dxb) <= Fb1.x) & (fabsf(dyb) <= Fb1.y) & (fabsf(dzb) <= Fb1.z);
    float qb = Fb1.w * dxb * dxb + Fb2.z * dyb * dyb + Fb3.x * dzb * dzb +
               2.f * (Fb2.x * dxb * dyb + Fb2.y * dxb * dzb + Fb2.w * dyb * dzb);
    float wb = (inB ? Fb0.w : 0.f) * __expf(-0.5f * qb);

    v2f a;
    a.x = wa;
    a.y = wb;

    // B operand: {feats[idx0][c], feats[idx1][c]} for c = m and m+16
    const float* f0 = feats + ip.x * CPAD + m;
    const float* f1 = feats + ip.y * CPAD + m;
    v2f b0, b1;
    b0.x = f0[0];  b0.y = f1[0];
    b1.x = f0[16]; b1.y = f1[16];

    acc0 = __builtin_amdgcn_wmma_f32_16x16x4_f32(false, a, false, b0, (short)0, acc0,
                                                 false, false);
    acc1 = __builtin_amdgcn_wmma_f32_16x16x4_f32(false, a, false, b1, (short)0, acc1,
                                                 false, false);
  }

  // C/D layout: VGPR i, lanes 0-15 -> M=i (N=lane), lanes 16-31 -> M=i+8
  float* gf = out + PTS;  // grid_feats region
#pragma unroll
  for (int i = 0; i < 8; ++i) {
    const int pp = tileBase + i + hl * 8;
    gf[pp * NCLS + m] = acc0[i];
    if (m < 2) gf[pp * NCLS + 16 + m] = acc1[i];
  }

  // grid_density is all zeros
  if (threadIdx.x < 128) out[B0 + threadIdx.x] = 0.f;
}

extern "C" void kernel_launch(void* const* d_in, const int* in_sizes, int n_in,
                              void* d_out, int out_size, void* d_ws, size_t ws_size,
                              hipStream_t stream) {
  (void)in_sizes; (void)n_in; (void)out_size; (void)ws_size;

  const float* means3d = (const float*)d_in[0];
  const float* opac    = (const float*)d_in[1];
  const float* scales  = (const float*)d_in[2];
  const float* rots    = (const float*)d_in[3];
  const float* feat    = (const float*)d_in[4];
  const float* empt    = (const float*)d_in[5];
  float* out = (float*)d_out;

  float* params   = (float*)d_ws;                 // NPAD*16 floats
  float* featsPad = params + NPAD * PSTRIDE;      // NPAD*32 floats

  gv_prep<<<(NPAD + 255) / 256, 256, 0, stream>>>(means3d, opac, scales, rots, feat,
                                                  empt, params, featsPad);

  gv_main<<<PTS / 128, 256, 0, stream>>>(params, featsPad, out);
}